// KGATLayer_46076409152044
// MI455X (gfx1250) — compile-verified
//
#include <hip/hip_runtime.h>
#include <hip/hip_bf16.h>
#include <math.h>

#define B_ 16
#define N_ 512
#define E_ 8192
#define D_ 256
#define LN_EPS 1e-5f
#define NEG_INF -1e9f

typedef float v2f __attribute__((ext_vector_type(2)));
typedef float v8f __attribute__((ext_vector_type(8)));

// ---------------------------------------------------------------------------
// K0: zero the scatter accumulator (must happen every call; graph replays).
// ---------------------------------------------------------------------------
__global__ void zero_ws(float* __restrict__ p, int n) {
    int i = blockIdx.x * blockDim.x + threadIdx.x;
    if (i < n) p[i] = 0.0f;
}

// ---------------------------------------------------------------------------
// K1: v_edge[i] = sum_o w_edge[o, i] * w_attn[513 + o]
// (te only appears via te . w_attn[513:769]; fold the whole edge GEMM here)
// ---------------------------------------------------------------------------
__global__ void edge_vec(const float* __restrict__ w_edge,
                         const float* __restrict__ w_attn,
                         float* __restrict__ v_edge) {
    int i = threadIdx.x;  // 256
    float acc = 0.0f;
    for (int o = 0; o < D_; ++o)
        acc = fmaf(w_edge[o * D_ + i], w_attn[2 * D_ + 1 + o], acc);
    v_edge[i] = acc;
}

// ---------------------------------------------------------------------------
// K2: tn = obj_nodes @ w_node^T via V_WMMA_F32_16X16X4_F32.
// One wave per 16x16 output tile; 64 chained K=4 accumulations.
// A tile: obj rows (M=16,K=4). B tile: B[k][o] = w_node[o][k] (contig float2).
// EXEC all ones: grid exactly covers tiles, no divergence.
// ---------------------------------------------------------------------------
__global__ void node_fc_wmma(const float* __restrict__ obj,
                             const float* __restrict__ w_node,
                             float* __restrict__ tn) {
    const int wave = blockIdx.x * (blockDim.x >> 5) + (threadIdx.x >> 5);
    const int lane = threadIdx.x & 31;
    const int tiles_per_b = (N_ / 16) * (D_ / 16);   // 32*16 = 512
    const int b   = wave / tiles_per_b;
    const int rem = wave % tiles_per_b;
    const int m0  = (rem / (D_ / 16)) * 16;          // node-row tile
    const int o0  = (rem % (D_ / 16)) * 16;          // out-feature tile
    const int half = lane >> 4;                      // 0: K={0,1}, 1: K={2,3}
    const int l15  = lane & 15;

    const float* arow = obj    + ((size_t)(b * N_ + m0 + l15)) * D_ + 2 * half;
    const float* brow = w_node + ((size_t)(o0 + l15)) * D_ + 2 * half;

    v8f c = {};
    #pragma unroll 8
    for (int k = 0; k < D_; k += 4) {
        v2f a  = *(const v2f*)(arow + k);
        v2f bm = *(const v2f*)(brow + k);
        c = __builtin_amdgcn_wmma_f32_16x16x4_f32(
                /*neg_a=*/false, a, /*neg_b=*/false, bm,
                /*c_mod=*/(short)0, c, /*reuse_a=*/false, /*reuse_b=*/false);
    }
    // C/D layout: VGPR r -> (M=r, N=lane) lanes 0-15 ; (M=8+r, N=lane-16) lanes 16-31
    float* dst = tn + ((size_t)(b * N_ + m0 + 8 * half)) * D_ + o0 + l15;
    #pragma unroll
    for (int r = 0; r < 8; ++r) dst[r * D_] = c[r];
}

// ---------------------------------------------------------------------------
// K3: per-node score contributions s1 = tn . wa[0:256], s2 = tn . wa[256:512]
// ---------------------------------------------------------------------------
__global__ void row_dots(const float* __restrict__ tn,
                         const float* __restrict__ w_attn,
                         float* __restrict__ s1, float* __restrict__ s2) {
    const int row = blockIdx.x;   // B*N
    const int t = threadIdx.x;    // 256
    float v = tn[(size_t)row * D_ + t];
    __shared__ float r1[256], r2[256];
    r1[t] = v * w_attn[t];
    r2[t] = v * w_attn[D_ + t];
    __syncthreads();
    for (int s = 128; s > 0; s >>= 1) {
        if (t < s) { r1[t] += r1[t + s]; r2[t] += r2[t + s]; }
        __syncthreads();
    }
    if (t == 0) { s1[row] = r1[0]; s2[row] = r2[0]; }
}

// ---------------------------------------------------------------------------
// K4: per-edge score. One wave per edge; float4 streaming of pred_emb (134MB
// -> the bandwidth-bound phase). leaky_relu then edge-mask -> NEG_INF.
// ---------------------------------------------------------------------------
__global__ void edge_scores(const float* __restrict__ pred,
                            const int* __restrict__ rel,
                            const float* __restrict__ sim,
                            const unsigned char* __restrict__ emask,
                            const float* __restrict__ s1,
                            const float* __restrict__ s2,
                            const float* __restrict__ v_edge,
                            const float* __restrict__ w_attn,
                            float* __restrict__ scores) {
    const int widx = blockIdx.x * 8 + (threadIdx.x >> 5);  // global edge (b*E+e)
    const int lane = threadIdx.x & 31;
    const int b = widx / E_;

    const float4* p = (const float4*)(pred + (size_t)widx * D_) + lane * 2;
    const float4* v = (const float4*)(v_edge) + lane * 2;
    float4 pa = p[0], pb = p[1];
    float4 va = v[0], vb = v[1];
    float dot = pa.x * va.x + pa.y * va.y + pa.z * va.z + pa.w * va.w
              + pb.x * vb.x + pb.y * vb.y + pb.z * vb.z + pb.w * vb.w;
    #pragma unroll
    for (int off = 16; off > 0; off >>= 1) dot += __shfl_xor(dot, off, 32);

    if (lane == 0) {
        float sc;
        if (emask[widx]) {
            int src = rel[widx * 2 + 0];
            int dst = rel[widx * 2 + 1];
            float x = s1[b * N_ + src] + s2[b * N_ + dst]
                    + sim[widx] * w_attn[2 * D_] + dot;
            sc = (x >= 0.0f) ? x : 0.01f * x;   // leaky_relu, slope 0.01
        } else {
            sc = NEG_INF;
        }
        scores[widx] = sc;
    }
}

// ---------------------------------------------------------------------------
// K5: per-batch softmax statistics over E edges (max + sum of exp).
// ---------------------------------------------------------------------------
__global__ void softmax_stats(const float* __restrict__ scores,
                              float* __restrict__ maxv, float* __restrict__ sumv) {
    const int b = blockIdx.x;
    const int t = threadIdx.x;   // 1024
    __shared__ float red[1024];
    float m = NEG_INF;
    for (int e = t; e < E_; e += 1024) m = fmaxf(m, scores[b * E_ + e]);
    red[t] = m; __syncthreads();
    for (int s = 512; s > 0; s >>= 1) {
        if (t < s) red[t] = fmaxf(red[t], red[t + s]);
        __syncthreads();
    }
    float mx = red[0]; __syncthreads();
    float sum = 0.0f;
    for (int e = t; e < E_; e += 1024) sum += expf(scores[b * E_ + e] - mx);
    red[t] = sum; __syncthreads();
    for (int s = 512; s > 0; s >>= 1) {
        if (t < s) red[t] += red[t + s];
        __syncthreads();
    }
    if (t == 0) { maxv[b] = mx; sumv[b] = red[0]; }
}

// ---------------------------------------------------------------------------
// K6: scatter messages: accum[b,dst,:] += weight(e) * tn[b,src,:].
// Masked edges carry weight exp(-1e9-max)=0 -> skipped entirely.
// ---------------------------------------------------------------------------
__global__ void scatter_msgs(const float* __restrict__ scores,
                             const float* __restrict__ maxv,
                             const float* __restrict__ sumv,
                             const int* __restrict__ rel,
                             const unsigned char* __restrict__ emask,
                             const float* __restrict__ tn,
                             float* __restrict__ accum) {
    const int widx = blockIdx.x * 8 + (threadIdx.x >> 5);
    const int lane = threadIdx.x & 31;
    const int b = widx / E_;
    if (!emask[widx]) return;
    float w = expf(scores[widx] - maxv[b]) / sumv[b];
    int src = rel[widx * 2 + 0];
    int dst = rel[widx * 2 + 1];
    const float* sp = tn    + ((size_t)(b * N_ + src)) * D_ + lane * 8;
    float*       dp = accum + ((size_t)(b * N_ + dst)) * D_ + lane * 8;
    #pragma unroll
    for (int j = 0; j < 8; ++j) unsafeAtomicAdd(dp + j, w * sp[j]);
}

// ---------------------------------------------------------------------------
// K7: node mask + LayerNorm over D, write d_out.
// ---------------------------------------------------------------------------
__global__ void finalize_ln(const float* __restrict__ accum,
                            const unsigned char* __restrict__ nmask,
                            const float* __restrict__ gamma,
                            const float* __restrict__ beta,
                            float* __restrict__ out) {
    const int row = blockIdx.x;  // B*N
    const int t = threadIdx.x;   // 256
    float x = accum[(size_t)row * D_ + t] * (nmask[row] ? 1.0f : 0.0f);
    __shared__ float red[256];
    red[t] = x; __syncthreads();
    for (int s = 128; s > 0; s >>= 1) {
        if (t < s) red[t] += red[t + s];
        __syncthreads();
    }
    float mu = red[0] * (1.0f / D_); __syncthreads();
    float dx = x - mu;
    red[t] = dx * dx; __syncthreads();
    for (int s = 128; s > 0; s >>= 1) {
        if (t < s) red[t] += red[t + s];
        __syncthreads();
    }
    float var = red[0] * (1.0f / D_);
    out[(size_t)row * D_ + t] = dx * rsqrtf(var + LN_EPS) * gamma[t] + beta[t];
}

// ---------------------------------------------------------------------------
extern "C" void kernel_launch(void* const* d_in, const int* in_sizes, int n_in,
                              void* d_out, int out_size, void* d_ws, size_t ws_size,
                              hipStream_t stream) {
    const float*         obj    = (const float*)d_in[0];
    const float*         pred   = (const float*)d_in[1];
    const int*           rel    = (const int*)d_in[2];
    const float*         sim    = (const float*)d_in[3];
    const unsigned char* nmask  = (const unsigned char*)d_in[4];  // bool mask
    const unsigned char* emask  = (const unsigned char*)d_in[5];  // bool mask
    const float*         w_node = (const float*)d_in[6];
    const float*         w_edge = (const float*)d_in[7];
    const float*         w_attn = (const float*)d_in[8];
    const float*         gamma  = (const float*)d_in[9];
    const float*         beta   = (const float*)d_in[10];
    float* out = (float*)d_out;

    float* ws     = (float*)d_ws;
    float* tn     = ws;                                  // B*N*D
    float* accum  = tn + (size_t)B_ * N_ * D_;           // B*N*D
    float* s1     = accum + (size_t)B_ * N_ * D_;        // B*N
    float* s2     = s1 + B_ * N_;                        // B*N
    float* scores = s2 + B_ * N_;                        // B*E
    float* v_edge = scores + B_ * E_;                    // D
    float* maxv   = v_edge + D_;                         // B
    float* sumv   = maxv + B_;                           // B

    const int bnd = B_ * N_ * D_;
    zero_ws<<<(bnd + 255) / 256, 256, 0, stream>>>(accum, bnd);
    edge_vec<<<1, 256, 0, stream>>>(w_edge, w_attn, v_edge);
    node_fc_wmma<<<(B_ * (N_ / 16) * (D_ / 16)) / 8, 256, 0, stream>>>(obj, w_node, tn);
    row_dots<<<B_ * N_, 256, 0, stream>>>(tn, w_attn, s1, s2);
    edge_scores<<<B_ * E_ / 8, 256, 0, stream>>>(pred, rel, sim, emask, s1, s2,
                                                 v_edge, w_attn, scores);
    softmax_stats<<<B_, 1024, 0, stream>>>(scores, maxv, sumv);
    scatter_msgs<<<B_ * E_ / 8, 256, 0, stream>>>(scores, maxv, sumv, rel, emask,
                                                  tn, accum);
    finalize_ln<<<B_ * N_, 256, 0, stream>>>(accum, nmask, gamma, beta, out);
}